// AutoRegMLAttention_28845000360008
// MI455X (gfx1250) — compile-verified
//
#include <hip/hip_runtime.h>
#include <math.h>

// Problem constants (from reference)
#define B_    16
#define L_    2048
#define HID_  4096
#define CQ_   1536
#define CKV_  512
#define HDR_  2048      // H*DR
#define HDH_  4096      // H*DH
#define LP1_  (L_ + 1)

typedef __attribute__((ext_vector_type(2))) float v2f;
typedef __attribute__((ext_vector_type(4))) float f4;
typedef __attribute__((ext_vector_type(8))) float v8f;

// ---------------------------------------------------------------------------
// WMMA f32 GEMM: C[16,N] = A[16,K] * B,   M == 16 == one WMMA tile.
//   BT=0: B is row-major [K,N]
//   BT=1: B is row-major [N,K]   (logical B[k,n] = Bmem[n*K + k])
// One wave computes one 16x16 C tile; K-loop in steps of 4 using
// V_WMMA_F32_16X16X4_F32. Pointer-increment addressing + unroll so the inner
// loop is loads + wmma, not 64-bit address rebuilds.
// A layout (ISA 7.12.2, 32-bit A 16x4): lanes 0-15 carry M=lane, K=k+{0,1};
// lanes 16-31 carry M=lane-16, K=k+{2,3}. B mirrors with N across lanes.
// C/D layout: VGPR i -> M = i + 8*(lane>=16), N = lane&15.
// Grids are sized to exactly cover N (all N are multiples of 64), so EXEC
// stays all-ones as WMMA requires.
// ---------------------------------------------------------------------------
template <int BT>
__global__ void wmma_gemm16(const float* __restrict__ A,
                            const float* __restrict__ Bm,
                            float* __restrict__ C, int K, int N) {
  const int lane  = threadIdx.x & 31;
  const int wave  = threadIdx.x >> 5;
  const int ntile = (blockIdx.x * (blockDim.x >> 5) + wave) * 16;
  const int half  = lane >> 4;            // 0: lanes 0-15, 1: lanes 16-31
  const int lid   = lane & 15;
  const int kofs  = half * 2;             // upper half supplies K+2, K+3

  const float* ap  = A + (size_t)lid * K + kofs;           // 8B-aligned (K even)
  const float* bp0;
  const float* bp1;
  if (BT) {
    bp0 = Bm + (size_t)(ntile + lid) * K + kofs;           // contiguous K-pair
    bp1 = bp0;                                             // unused
  } else {
    bp0 = Bm + (size_t)kofs * N + ntile + lid;             // row k+kofs
    bp1 = bp0 + N;                                         // row k+kofs+1
  }
  const size_t bstep = BT ? 4 : 4 * (size_t)N;

  v8f c = {};
#pragma unroll 4
  for (int k = 0; k < K; k += 4) {
    v2f a = *(const v2f*)ap;  ap += 4;
    v2f b;
    if (BT) {
      b = *(const v2f*)bp0;
    } else {
      b.x = *bp0;
      b.y = *bp1;
      bp1 += bstep;
    }
    bp0 += bstep;
    // (neg_a, A, neg_b, B, c_mod, C, reuse_a, reuse_b)
    c = __builtin_amdgcn_wmma_f32_16x16x4_f32(false, a, false, b,
                                              (short)0, c, false, false);
  }
#pragma unroll
  for (int i = 0; i < 8; ++i)
    C[(size_t)(i + half * 8) * N + ntile + lid] = c[i];
}

// ---------------------------------------------------------------------------
// Prefix attention scores:
//   scores[b,l] = ( qCeff[b]·cKV[b,l] + qR[b]·kR[b,l] ) / sqrt(192)
// q vectors staged in LDS; one wave per cache row, float4 coalesced loads,
// wave32 shuffle reduction. kR cache (268 MB, single-use) is loaded
// non-temporally so the reusable cKV cache (67 MB) stays resident in L2.
// ---------------------------------------------------------------------------
__global__ void mla_prefix_scores(const float* __restrict__ qCeff,
                                  const float* __restrict__ qR,
                                  const float* __restrict__ cKVc,
                                  const float* __restrict__ kRc,
                                  float* __restrict__ scores) {
  const int b    = blockIdx.x;
  const int lane = threadIdx.x & 31;
  const int wave = threadIdx.x >> 5;
  __shared__ float qc[CKV_];
  __shared__ float qr[HDR_];
  for (int i = threadIdx.x; i < CKV_; i += blockDim.x) qc[i] = qCeff[b * CKV_ + i];
  for (int i = threadIdx.x; i < HDR_; i += blockDim.x) qr[i] = qR[b * HDR_ + i];
  __syncthreads();

  const float rs = rsqrtf(192.0f);
  const int rowsPerBlock = L_ / gridDim.y;
  const int rowsPerWave  = rowsPerBlock / (blockDim.x >> 5);
  const int l0 = blockIdx.y * rowsPerBlock + wave * rowsPerWave;

  for (int r = 0; r < rowsPerWave; ++r) {
    const int l = l0 + r;
    const f4* kv = (const f4*)(cKVc + ((size_t)b * L_ + l) * CKV_);
    const f4* kr = (const f4*)(kRc  + ((size_t)b * L_ + l) * HDR_);
    float acc = 0.0f;
    for (int i = lane; i < CKV_ / 4; i += 32) {
      f4 v = kv[i];
      acc += v.x * qc[4 * i] + v.y * qc[4 * i + 1] + v.z * qc[4 * i + 2] + v.w * qc[4 * i + 3];
    }
    for (int i = lane; i < HDR_ / 4; i += 32) {
      f4 v = __builtin_nontemporal_load(&kr[i]);
      acc += v.x * qr[4 * i] + v.y * qr[4 * i + 1] + v.z * qr[4 * i + 2] + v.w * qr[4 * i + 3];
    }
    for (int off = 16; off > 0; off >>= 1) acc += __shfl_down(acc, off, 32);
    if (lane == 0) scores[(size_t)b * LP1_ + l] = acc * rs;
  }
}

// New-token score: scores[b, L] = (qCeff[b]·cKV_new[b] + qR[b]·kR_new[b]) / sqrt(192)
// 16 waves, one per batch.
__global__ void mla_new_score(const float* __restrict__ qCeff,
                              const float* __restrict__ qR,
                              const float* __restrict__ cKVn,
                              const float* __restrict__ kRn,
                              float* __restrict__ scores) {
  const int lane = threadIdx.x & 31;
  const int b    = threadIdx.x >> 5;
  float acc = 0.0f;
  for (int i = lane; i < CKV_; i += 32) acc += qCeff[b * CKV_ + i] * cKVn[b * CKV_ + i];
  for (int i = lane; i < HDR_; i += 32) acc += qR[b * HDR_ + i] * kRn[b * HDR_ + i];
  for (int off = 16; off > 0; off >>= 1) acc += __shfl_down(acc, off, 32);
  if (lane == 0) scores[(size_t)b * LP1_ + L_] = acc * rsqrtf(192.0f);
}

// Softmax over L+1 scores per batch.
__global__ void mla_softmax(const float* __restrict__ scores,
                            float* __restrict__ probs) {
  const int b = blockIdx.x;
  __shared__ float red[256];
  const float* s = scores + (size_t)b * LP1_;
  float m = -INFINITY;
  for (int i = threadIdx.x; i < LP1_; i += blockDim.x) m = fmaxf(m, s[i]);
  red[threadIdx.x] = m; __syncthreads();
  for (int st = 128; st > 0; st >>= 1) {
    if (threadIdx.x < st) red[threadIdx.x] = fmaxf(red[threadIdx.x], red[threadIdx.x + st]);
    __syncthreads();
  }
  m = red[0]; __syncthreads();
  float sum = 0.0f;
  for (int i = threadIdx.x; i < LP1_; i += blockDim.x) sum += expf(s[i] - m);
  red[threadIdx.x] = sum; __syncthreads();
  for (int st = 128; st > 0; st >>= 1) {
    if (threadIdx.x < st) red[threadIdx.x] += red[threadIdx.x + st];
    __syncthreads();
  }
  const float inv = 1.0f / red[0];
  for (int i = threadIdx.x; i < LP1_; i += blockDim.x)
    probs[(size_t)b * LP1_ + i] = expf(s[i] - m) * inv;
}

// Partial weighted sum over an L-chunk: part[b,ch,k] = sum_l p[b,l]*cKV[b,l,k]
// threads span k (coalesced per l-row). Deterministic (no float atomics).
// cKV rows should be L2 hits after the score pass.
__global__ void mla_ctx_partial(const float* __restrict__ probs,
                                const float* __restrict__ cKVc,
                                float* __restrict__ part) {
  const int b  = blockIdx.x;
  const int ch = blockIdx.y;
  const int k  = threadIdx.x;                 // 512 threads
  const int lpc = L_ / gridDim.y;
  const float* pv = probs + (size_t)b * LP1_ + (size_t)ch * lpc;
  const float* kv = cKVc + ((size_t)b * L_ + (size_t)ch * lpc) * CKV_ + k;
  float acc = 0.0f;
  for (int l = 0; l < lpc; ++l) acc += pv[l] * kv[(size_t)l * CKV_];
  part[((size_t)b * gridDim.y + ch) * CKV_ + k] = acc;
}

// ctx[b,k] = sum_ch part[b,ch,k] + p[b,L]*cKV_new[b,k]
__global__ void mla_ctx_reduce(const float* __restrict__ part,
                               const float* __restrict__ probs,
                               const float* __restrict__ cKVn,
                               float* __restrict__ ctx, int nch) {
  const int b = blockIdx.x;
  const int k = threadIdx.x;
  float acc = probs[(size_t)b * LP1_ + L_] * cKVn[b * CKV_ + k];
  for (int c = 0; c < nch; ++c) acc += part[((size_t)b * nch + c) * CKV_ + k];
  ctx[b * CKV_ + k] = acc;
}

// Cache concat: dst[b, 0..L-1, :] = src[b], dst[b, L, :] = newrow[b].
// One block per destination row -> scalar div only once per block; float4
// nontemporal streaming (this is the dominant 670 MB of traffic).
__global__ void concat_cache(const float* __restrict__ src,
                             const float* __restrict__ newrow,
                             float* __restrict__ dst, int D4) {
  const int row = blockIdx.x;                 // B*(L+1) rows
  const int b   = row / LP1_;
  const int r   = row - b * LP1_;
  const f4* s = (r < L_)
      ? (const f4*)src + ((size_t)b * L_ + r) * D4
      : (const f4*)newrow + (size_t)b * D4;
  f4* d = (f4*)dst + (size_t)row * D4;
  for (int c = threadIdx.x; c < D4; c += blockDim.x) {
    f4 v = __builtin_nontemporal_load(&s[c]);
    __builtin_nontemporal_store(v, &d[c]);
  }
}

// ---------------------------------------------------------------------------
extern "C" void kernel_launch(void* const* d_in, const int* in_sizes, int n_in,
                              void* d_out, int out_size, void* d_ws, size_t ws_size,
                              hipStream_t stream) {
  (void)in_sizes; (void)n_in; (void)out_size; (void)ws_size;
  const float* hs    = (const float*)d_in[0];   // [B,1,HID]
  const float* cKVc  = (const float*)d_in[1];   // [B,L,CKV]
  const float* kRc   = (const float*)d_in[2];   // [B,L,H*DR]
  const float* W_DQ  = (const float*)d_in[3];   // [HID,CQ]
  const float* W_DKV = (const float*)d_in[4];   // [HID,CKV]
  const float* W_UQC = (const float*)d_in[5];   // [CQ,H*DH]
  const float* W_UQR = (const float*)d_in[6];   // [CQ,H*DR]
  const float* W_KR  = (const float*)d_in[7];   // [HID,H*DR]
  const float* W_UKC = (const float*)d_in[8];   // [CKV,H*DH]
  const float* W_UVC = (const float*)d_in[9];   // [CKV,H*DH]
  const float* W_O   = (const float*)d_in[10];  // [H*DH,HID]

  float* out     = (float*)d_out;                               // [B,1,HID]
  float* out_ckv = out + (size_t)B_ * HID_;                     // [B,L+1,CKV]
  float* out_kr  = out_ckv + (size_t)B_ * LP1_ * CKV_;          // [B,L+1,H*DR]

  float* ws     = (float*)d_ws;
  float* cQ     = ws; ws += B_ * CQ_;
  float* qC     = ws; ws += B_ * HDH_;
  float* qR     = ws; ws += B_ * HDR_;
  float* kR     = ws; ws += B_ * HDR_;
  float* cKVn   = ws; ws += B_ * CKV_;
  float* qCeff  = ws; ws += B_ * CKV_;
  float* scores = ws; ws += B_ * LP1_;
  float* probs  = ws; ws += B_ * LP1_;
  float* part   = ws; ws += B_ * 8 * CKV_;
  float* ctx    = ws; ws += B_ * CKV_;
  float* t2     = ws; ws += B_ * HDH_;

  // Dense projections — all M=16 WMMA-f32 GEMMs (4 waves/block, 16 N-cols each)
  wmma_gemm16<0><<<dim3(CQ_  / 64), 128, 0, stream>>>(hs, W_DQ,  cQ,    HID_, CQ_);
  wmma_gemm16<0><<<dim3(CKV_ / 64), 128, 0, stream>>>(hs, W_DKV, cKVn,  HID_, CKV_);
  wmma_gemm16<0><<<dim3(HDR_ / 64), 128, 0, stream>>>(hs, W_KR,  kR,    HID_, HDR_);  // RoPE@pos0 == identity
  wmma_gemm16<0><<<dim3(HDH_ / 64), 128, 0, stream>>>(cQ, W_UQC, qC,    CQ_,  HDH_);
  wmma_gemm16<0><<<dim3(HDR_ / 64), 128, 0, stream>>>(cQ, W_UQR, qR,    CQ_,  HDR_);  // RoPE@pos0 == identity
  wmma_gemm16<1><<<dim3(CKV_ / 64), 128, 0, stream>>>(qC, W_UKC, qCeff, HDH_, CKV_);  // qC @ W_UK_C^T

  // Cache concatenation (ready once cKVn and kR exist) — dominant traffic
  concat_cache<<<B_ * LP1_, 128, 0, stream>>>(cKVc, cKVn, out_ckv, CKV_ / 4);
  concat_cache<<<B_ * LP1_, 256, 0, stream>>>(kRc,  kR,   out_kr,  HDR_ / 4);

  // Attention (no head dim: reference contracts all heads into one 2560-d score)
  mla_new_score<<<1, 512, 0, stream>>>(qCeff, qR, cKVn, kR, scores);
  mla_prefix_scores<<<dim3(B_, 32), 256, 0, stream>>>(qCeff, qR, cKVc, kRc, scores);
  mla_softmax<<<B_, 256, 0, stream>>>(scores, probs);
  mla_ctx_partial<<<dim3(B_, 8), CKV_, 0, stream>>>(probs, cKVc, part);
  mla_ctx_reduce<<<B_, CKV_, 0, stream>>>(part, probs, cKVn, ctx, 8);

  // output = (probs·[cKV_cache; cKV_new]) @ W_UV_C @ W_O   (reordered: avoids W_VO)
  wmma_gemm16<0><<<dim3(HDH_ / 64), 128, 0, stream>>>(ctx, W_UVC, t2, CKV_, HDH_);
  wmma_gemm16<0><<<dim3(HID_ / 64), 128, 0, stream>>>(t2,  W_O,  out, HDH_, HID_);
}